// ParietalCortex42PyTorch_87935160418590
// MI455X (gfx1250) — compile-verified
//
#include <hip/hip_runtime.h>
#include <hip/hip_bf16.h>
#include <stdint.h>

// MI455X / gfx1250, wave32.
//   C = where(eye, corr, 0.9*corr + 0.1*outer(a,a)); sums of |C|, |C|*W, W.
// HBM-bound: 768MB @ 23.3 TB/s => ~33us floor. The rank-1 update is expressed
// as a true GEMM per 16x16 tile via V_WMMA_F32_16X16X4_F32 (exact f32):
//   D = A(:,0)=a_i  x  B(0,:)=0.1*a_j  +  0.9*corr_tile
// Tiles are staged by the CDNA5 async DMA engine (GLOBAL_LOAD_ASYNC_TO_LDS_B128,
// ASYNCcnt-paced double buffering) and re-read from LDS in WMMA C/D layout
// (LDS is random access => no strided-global penalty). Output uses
// non-temporal stores so the dead 256MB stream doesn't thrash L2.

typedef float v2f __attribute__((ext_vector_type(2)));
typedef float v8f __attribute__((ext_vector_type(8)));

#define THREADS 256
#define BLOCKS  2048
#define NWAVES  (THREADS / 32)

// ---- CDNA5 async global->LDS DMA (ASYNCcnt-tracked) ------------------------
__device__ __forceinline__ void async_load_b128(uint32_t lds_byte_addr,
                                                const void* gaddr) {
  asm volatile("global_load_async_to_lds_b128 %0, %1, off"
               :: "v"(lds_byte_addr), "v"(gaddr)
               : "memory");
}
__device__ __forceinline__ void wait_async_le4() {
  asm volatile("s_wait_asynccnt 0x4" ::: "memory");
}
__device__ __forceinline__ void wait_async_0() {
  asm volatile("s_wait_asynccnt 0x0" ::: "memory");
}

// Stage one 16x16 f32 tile (row-major, 1KB) into LDS: 2 async b128 ops,
// 4 lanes cover each 64B row, 8 rows per op.
__device__ __forceinline__ void issue_tile(const float* __restrict__ base,
                                           unsigned n, unsigned tM, unsigned tN,
                                           uint32_t lds_tile_base, int lane) {
  const unsigned r = (unsigned)lane >> 2;      // 0..7
  const unsigned q = (unsigned)lane & 3;       // 16B quarter within row
  const float* g0 = base + (size_t)(tM * 16u + r) * n + (size_t)(tN * 16u + q * 4u);
  async_load_b128(lds_tile_base + (unsigned)lane * 16u,        g0);
  async_load_b128(lds_tile_base + (unsigned)lane * 16u + 512u, g0 + (size_t)8 * n);
}

// ---- main streaming kernel -------------------------------------------------
__global__ __launch_bounds__(THREADS)
void pc42_wmma_stream(const float* __restrict__ act,
                      const float* __restrict__ W,
                      const float* __restrict__ corr,
                      float* __restrict__ Cout,
                      float* __restrict__ partials,       // [3 * gridDim.x]
                      unsigned n, unsigned logTiles, unsigned maskTiles,
                      long long numTiles)
{
  __shared__ __align__(16) float s_c[NWAVES][2][256];   // corr tiles (dbl buf)
  __shared__ __align__(16) float s_w[NWAVES][2][256];   // W tiles

  const int tid  = threadIdx.x;
  const int lane = tid & 31;
  const int wave = tid >> 5;
  const unsigned hi = (unsigned)lane >> 4;              // lane half: 0/1
  const unsigned nl = (unsigned)lane & 15u;             // N position

  const long long waveStride = (long long)gridDim.x * NWAVES;
  const long long t0 = (long long)blockIdx.x * NWAVES + wave;  // wave-uniform

  uint32_t lc[2], lw[2];                                // LDS tile base addrs
  lc[0] = (uint32_t)(uintptr_t)&s_c[wave][0][0];
  lc[1] = (uint32_t)(uintptr_t)&s_c[wave][1][0];
  lw[0] = (uint32_t)(uintptr_t)&s_w[wave][0][0];
  lw[1] = (uint32_t)(uintptr_t)&s_w[wave][1][0];

  // Prologue: stage two tiles (clamped dummies keep ASYNCcnt uniform).
  {
    long long tA = (t0 < numTiles) ? t0 : 0;
    long long tB = (t0 + waveStride < numTiles) ? (t0 + waveStride) : 0;
    unsigned aM = (unsigned)(tA >> logTiles), aN = (unsigned)tA & maskTiles;
    unsigned bM = (unsigned)(tB >> logTiles), bN = (unsigned)tB & maskTiles;
    issue_tile(corr, n, aM, aN, lc[0], lane);
    issue_tile(W,    n, aM, aN, lw[0], lane);
    issue_tile(corr, n, bM, bN, lc[1], lane);
    issue_tile(W,    n, bM, bN, lw[1], lane);
  }

  float sAbs = 0.0f, sAW = 0.0f, sW = 0.0f;
  int buf = 0;
  for (long long t = t0; t < numTiles; t += waveStride, buf ^= 1) {
    const unsigned tM = (unsigned)(t >> logTiles);
    const unsigned tN = (unsigned)t & maskTiles;

    wait_async_le4();               // current tile's 4 DMAs have landed

    // Read raw tiles from LDS in WMMA C/D layout: slot v <-> (M=v+8*hi, N=nl)
    float rawc[8], wv[8];
    #pragma unroll
    for (int v = 0; v < 8; ++v) {
      rawc[v] = s_c[wave][buf][(unsigned)(v + 8 * hi) * 16u + nl];
      wv[v]   = s_w[wave][buf][(unsigned)(v + 8 * hi) * 16u + nl];
    }

    // Refill this buffer for tile t + 2*waveStride (dummy-clamped at tail).
    long long tp = t + 2 * waveStride;
    if (tp >= numTiles) tp = 0;
    {
      unsigned pM = (unsigned)(tp >> logTiles), pN = (unsigned)tp & maskTiles;
      issue_tile(corr, n, pM, pN, lc[buf], lane);
      issue_tile(W,    n, pM, pN, lw[buf], lane);
    }

    // Rank-1 GEMM via WMMA: A(:,K=0) = a_i, B(K=0,:) = 0.1*a_j, C = 0.9*corr.
    // A layout (16x4 f32): VGPR0 = K0 (lanes 0-15) / K2 (lanes 16-31), VGPR1 = K1/K3.
    // Zero everything except the K=0 slots in lanes 0-15 of VGPR0.
    const float av = act[(size_t)tM * 16u + nl];
    const float bv = act[(size_t)tN * 16u + nl];
    v2f a, b;
    a.x = hi ? 0.0f : av;          a.y = 0.0f;
    b.x = hi ? 0.0f : 0.1f * bv;   b.y = 0.0f;
    v8f cin;
    #pragma unroll
    for (int v = 0; v < 8; ++v) cin[v] = 0.9f * rawc[v];

    v8f d = __builtin_amdgcn_wmma_f32_16x16x4_f32(
        /*neg_a=*/false, a, /*neg_b=*/false, b,
        /*c_mod=*/(short)0, cin, /*reuse_a=*/false, /*reuse_b=*/false);

    // Diagonal tiles keep original corr on the diagonal.
    if (tM == tN) {
      const int ve = (int)nl - (int)(hi << 3);     // M==N  <=>  v == ve
      if (ve >= 0 && ve < 8) d[ve] = rawc[ve];
    }

    // Reductions + non-temporal b32 stores (two 64B segments per store).
    float* p = Cout + (size_t)(tM * 16u + hi * 8u) * n + (size_t)(tN * 16u + nl);
    #pragma unroll
    for (int v = 0; v < 8; ++v) {
      const float ad = __builtin_fabsf(d[v]);
      sAbs += ad;
      sAW  += ad * wv[v];
      sW   += wv[v];
      __builtin_nontemporal_store(d[v], p + (size_t)v * n);
    }
  }
  wait_async_0();                   // quiesce tail prefetches

  // wave32 reduction, then per-block partials (deterministic: no atomics)
  #pragma unroll
  for (int off = 16; off > 0; off >>= 1) {
    sAbs += __shfl_down(sAbs, off);
    sAW  += __shfl_down(sAW,  off);
    sW   += __shfl_down(sW,   off);
  }
  __shared__ float r0[NWAVES], r1[NWAVES], r2[NWAVES];
  if ((tid & 31) == 0) { r0[wave] = sAbs; r1[wave] = sAW; r2[wave] = sW; }
  __syncthreads();
  if (tid == 0) {
    float a0 = 0.f, a1 = 0.f, a2 = 0.f;
    #pragma unroll
    for (int i = 0; i < NWAVES; ++i) { a0 += r0[i]; a1 += r1[i]; a2 += r2[i]; }
    partials[blockIdx.x]                 = a0;
    partials[gridDim.x + blockIdx.x]     = a1;
    partials[2 * gridDim.x + blockIdx.x] = a2;
  }
}

// ---- fold partials + emit the three scalars --------------------------------
__global__ __launch_bounds__(256)
void pc42_finalize(const float* __restrict__ partials, int nblocks,
                   const float* __restrict__ cs, const float* __restrict__ im,
                   float* __restrict__ scal, long long n)
{
  const int tid = threadIdx.x;
  float t0 = 0.f, t1 = 0.f, t2 = 0.f;
  for (int i = tid; i < nblocks; i += 256) {
    t0 += partials[i];
    t1 += partials[nblocks + i];
    t2 += partials[2 * nblocks + i];
  }
  #pragma unroll
  for (int off = 16; off > 0; off >>= 1) {
    t0 += __shfl_down(t0, off);
    t1 += __shfl_down(t1, off);
    t2 += __shfl_down(t2, off);
  }
  __shared__ float r0[8], r1[8], r2[8];
  const int lane = tid & 31, wave = tid >> 5;
  if (lane == 0) { r0[wave] = t0; r1[wave] = t1; r2[wave] = t2; }
  __syncthreads();
  if (tid == 0) {
    float a = 0.f, b = 0.f, c = 0.f;
    #pragma unroll
    for (int i = 0; i < 8; ++i) { a += r0[i]; b += r1[i]; c += r2[i]; }
    const float fn = (float)n;
    const float nconn = fn * (fn - 1.0f);
    scal[0] = 0.9f * cs[0] + 0.1f * (a / nconn);                    // coherence
    scal[1] = (c > 0.0f) ? (0.9f * im[0] + 0.1f * (b / c)) : im[0]; // integration
    scal[2] = a / (fn * fn);                                        // synchrony
  }
}

// ---- harness entry ---------------------------------------------------------
extern "C" void kernel_launch(void* const* d_in, const int* in_sizes, int n_in,
                              void* d_out, int out_size, void* d_ws, size_t ws_size,
                              hipStream_t stream) {
  const float* act  = (const float*)d_in[0];
  const float* W    = (const float*)d_in[1];
  const float* corr = (const float*)d_in[2];
  const float* cs   = (const float*)d_in[3];
  const float* im   = (const float*)d_in[4];

  const long long n     = in_sizes[0];           // 8192 (power of two)
  const long long total = n * n;

  unsigned logn = 0;
  while ((1ll << logn) < n) logn++;
  const unsigned logTiles  = logn - 4;           // tiles per row = n/16
  const unsigned maskTiles = (unsigned)((n >> 4) - 1);
  const long long numTiles = (n >> 4) * (n >> 4);

  float* C        = (float*)d_out;
  float* scal     = C + total;                   // coherence, integration, synchrony
  float* partials = (float*)d_ws;                // 3 * BLOCKS floats (24 KB)

  pc42_wmma_stream<<<dim3(BLOCKS), dim3(THREADS), 0, stream>>>(
      act, W, corr, C, partials, (unsigned)n, logTiles, maskTiles, numTiles);
  pc42_finalize<<<dim3(1), dim3(256), 0, stream>>>(
      partials, BLOCKS, cs, im, scal, n);
}